// _MambaBlock_82471962018365
// MI455X (gfx1250) — compile-verified
//
#include <hip/hip_runtime.h>

typedef __attribute__((ext_vector_type(2))) float v2f;
typedef __attribute__((ext_vector_type(8))) float v8f;

constexpr int DM  = 768;         // d_model
constexpr int DI  = 1536;        // d_inner
constexpr int DS  = 16;          // d_state
constexpr int DC  = 4;           // d_conv
constexpr int L   = 2048;
constexpr int NXZ = 2 * DI;          // 3072
constexpr int NXP = DI + 2 * DS;     // 1568
constexpr int NXP_PAD = 1600;        // padded to multiple of 64 for GEMM tiles

// ---------------------------------------------------------------------------
// CDNA5 async-to-LDS helpers (ASYNCcnt-tracked, in-order completion)
// ---------------------------------------------------------------------------
__device__ __forceinline__ unsigned lds_lo32(const void* p) {
  // generic pointers to LDS carry the byte offset in the low 32 bits
  return (unsigned)(uintptr_t)p;
}
__device__ __forceinline__ void async_load_b128(unsigned lds, const float* gaddr) {
  asm volatile("global_load_async_to_lds_b128 %0, %1, off"
               :: "v"(lds), "v"(gaddr) : "memory");
}
__device__ __forceinline__ void wait_async0() {
  asm volatile("s_wait_asynccnt 0x0" ::: "memory");
}
__device__ __forceinline__ void wait_async3() {
  asm volatile("s_wait_asynccnt 0x3" ::: "memory");
}

// ---------------------------------------------------------------------------
// LayerNorm: one block per row (768 elems), wave32 shuffle + LDS reduction
// ---------------------------------------------------------------------------
__global__ __launch_bounds__(256) void ln_kernel(const float* __restrict__ x,
                                                 const float* __restrict__ gamma,
                                                 const float* __restrict__ beta,
                                                 float* __restrict__ xn) {
  const int row = blockIdx.x;
  const float* xr = x + (size_t)row * DM;
  float s = 0.f, ss = 0.f;
  for (int i = threadIdx.x; i < DM; i += 256) { float v = xr[i]; s += v; ss += v * v; }
  for (int off = 16; off > 0; off >>= 1) {
    s  += __shfl_xor(s,  off, 32);
    ss += __shfl_xor(ss, off, 32);
  }
  __shared__ float red0[8], red1[8];
  __shared__ float smu, srstd;
  const int wave = threadIdx.x >> 5, lane = threadIdx.x & 31;
  if (lane == 0) { red0[wave] = s; red1[wave] = ss; }
  __syncthreads();
  if (threadIdx.x == 0) {
    float S = 0.f, SS = 0.f;
    for (int w = 0; w < 8; ++w) { S += red0[w]; SS += red1[w]; }
    float m = S / (float)DM;
    smu = m;
    srstd = rsqrtf(SS / (float)DM - m * m + 1e-5f);
  }
  __syncthreads();
  const float m = smu, r = srstd;
  for (int i = threadIdx.x; i < DM; i += 256)
    xn[(size_t)row * DM + i] = (xr[i] - m) * r * gamma[i] + beta[i];
}

// ---------------------------------------------------------------------------
// fp32 WMMA GEMM: C[M,N] = A[M,K] * B[K,N] (+ optional residual)
// Block tile 128x64, 8 waves = 4(M) x 2(N); wave tile 32x32 -> 4 accumulators.
// K staged 16 at a time through double-buffered LDS filled by
// GLOBAL_LOAD_ASYNC_TO_LDS_B128 (3 per thread per stage); stage s+1 fetch
// overlaps WMMA of stage s.  Steady-state loop is branch-free (last stage
// peeled) with strided global-pointer increments; wait is asynccnt <= 3
// (async loads complete in order).
// LDS strides: A=20 / B=80 floats (bank-conflict-free fragment reads,
// 16B-aligned B128 destinations).  Requires M%128==0, N%64==0, K%32==0.
// ---------------------------------------------------------------------------
__global__ __launch_bounds__(256) void gemm_wmma_f32(const float* __restrict__ A,
                                                     const float* __restrict__ B,
                                                     float* __restrict__ C,
                                                     int M, int N, int K,
                                                     const float* __restrict__ resid,
                                                     int resid_ld) {
  __shared__ float As[2][128][20];
  __shared__ float Bs[2][16][80];

  const int bm = blockIdx.y * 128;
  const int bn = blockIdx.x * 64;
  const int tid  = threadIdx.x;
  const int wave = tid >> 5;
  const int lane = tid & 31;
  const int wm = (wave >> 1) * 32;   // 0,32,64,96
  const int wn = (wave & 1) * 32;    // 0,32

  v8f acc00 = {}, acc01 = {}, acc10 = {}, acc11 = {};

  // cooperative async-load mappings (256 threads)
  // A tile: 128 rows x 16 cols -> two B128 per thread (row = tid>>1, col 0/8)
  const int ar0 = tid >> 1;
  const int aq0 = (tid & 1) << 3;
  // B tile: 16 rows x 64 cols -> one B128 per thread
  const int br = tid >> 4, bq = (tid & 15) << 2;

  const int mrow  = lane & 15;
  const int khalf = (lane >> 4) << 1;       // 0 or 2 (ISA 16x4 f32 A/B layout)

  const int S = K >> 4;                     // #16-wide K stages (>= 2)

  // per-thread LDS destinations for both buffers
  const unsigned lA[2] = { lds_lo32(&As[0][ar0][aq0]), lds_lo32(&As[1][ar0][aq0]) };
  const unsigned lB[2] = { lds_lo32(&Bs[0][br][bq]),   lds_lo32(&Bs[1][br][bq])   };

  // strided global pointers (advance per stage by 16 floats / 16 rows)
  const float* ga = A + (size_t)(bm + ar0) * K + aq0;
  const float* gb = B + (size_t)br * N + bn + bq;
  const size_t gbstride = (size_t)16 * N;

  auto compute = [&](const float (*Asb)[20], const float (*Bsb)[80]) {
#pragma unroll
    for (int kk = 0; kk < 16; kk += 4) {
      v2f a0, a1, b0, b1;
      a0.x = Asb[wm + mrow][kk + khalf];
      a0.y = Asb[wm + mrow][kk + khalf + 1];
      a1.x = Asb[wm + 16 + mrow][kk + khalf];
      a1.y = Asb[wm + 16 + mrow][kk + khalf + 1];
      b0.x = Bsb[kk + khalf][wn + mrow];
      b0.y = Bsb[kk + khalf + 1][wn + mrow];
      b1.x = Bsb[kk + khalf][wn + 16 + mrow];
      b1.y = Bsb[kk + khalf + 1][wn + 16 + mrow];
      acc00 = __builtin_amdgcn_wmma_f32_16x16x4_f32(false, a0, false, b0,
                                                    (short)0, acc00, false, false);
      acc01 = __builtin_amdgcn_wmma_f32_16x16x4_f32(false, a0, false, b1,
                                                    (short)0, acc01, false, false);
      acc10 = __builtin_amdgcn_wmma_f32_16x16x4_f32(false, a1, false, b0,
                                                    (short)0, acc10, false, false);
      acc11 = __builtin_amdgcn_wmma_f32_16x16x4_f32(false, a1, false, b1,
                                                    (short)0, acc11, false, false);
    }
  };

  // prologue: stage 0 into buffer 0
  async_load_b128(lA[0],      ga);
  async_load_b128(lA[0] + 16, ga + 4);
  async_load_b128(lB[0],      gb);

  int buf = 0;
  for (int s = 0; s < S - 1; ++s) {
    ga += 16;
    gb += gbstride;
    const int nbuf = buf ^ 1;
    async_load_b128(lA[nbuf],      ga);
    async_load_b128(lA[nbuf] + 16, ga + 4);
    async_load_b128(lB[nbuf],      gb);
    wait_async3();                 // stage s complete (in-order), s+1 in flight
    __syncthreads();
    compute(As[buf], Bs[buf]);
    __syncthreads();               // all waves done with buf before refill
    buf = nbuf;
  }
  wait_async0();                   // last stage
  __syncthreads();
  compute(As[buf], Bs[buf]);

  // C layout (ISA 16x16 f32): lane n = lane%16, VGPR v -> m = v + (lane>=16)*8
  const int cn0 = bn + wn + (lane & 15);
  const int cn1 = cn0 + 16;
  const int mb  = bm + wm + ((lane >> 4) << 3);
#pragma unroll
  for (int v = 0; v < 8; ++v) {
    const int m0 = mb + v;
    const int m1 = mb + 16 + v;
    float c00 = acc00[v], c01 = acc01[v], c10 = acc10[v], c11 = acc11[v];
    if (resid) {
      c00 += resid[(size_t)m0 * resid_ld + cn0];
      c01 += resid[(size_t)m0 * resid_ld + cn1];
      c10 += resid[(size_t)m1 * resid_ld + cn0];
      c11 += resid[(size_t)m1 * resid_ld + cn1];
    }
    C[(size_t)m0 * N + cn0] = c00;
    C[(size_t)m0 * N + cn1] = c01;
    C[(size_t)m1 * N + cn0] = c10;
    C[(size_t)m1 * N + cn1] = c11;
  }
}

// ---------------------------------------------------------------------------
// Depthwise causal conv (k=4) + bias + SiLU. Reads xb slice of xz (cols 0..DI).
// ---------------------------------------------------------------------------
__global__ __launch_bounds__(256) void conv_silu_kernel(const float* __restrict__ xz,
                                                        const float* __restrict__ cw,
                                                        const float* __restrict__ cb,
                                                        float* __restrict__ xb) {
  const int idx = blockIdx.x * 256 + threadIdx.x;
  if (idx >= L * DI) return;
  const int t = idx / DI;
  const int d = idx - t * DI;
  float acc = cb[d];
#pragma unroll
  for (int j = 0; j < DC; ++j) {
    const int tt = t - (DC - 1) + j;
    if (tt >= 0) acc += cw[d * DC + j] * xz[(size_t)tt * NXZ + d];
  }
  xb[idx] = acc / (1.f + __expf(-acc));   // silu
}

// ---------------------------------------------------------------------------
// Zero-pad W_xproj (1536 x 1568) -> (1536 x 1600)
// ---------------------------------------------------------------------------
__global__ __launch_bounds__(256) void pad_wxproj(const float* __restrict__ w,
                                                  float* __restrict__ wp) {
  const int idx = blockIdx.x * 256 + threadIdx.x;
  if (idx >= DI * NXP_PAD) return;
  const int r = idx / NXP_PAD;
  const int c = idx - r * NXP_PAD;
  wp[idx] = (c < NXP) ? w[(size_t)r * NXP + c] : 0.f;
}

// ---------------------------------------------------------------------------
// Sequential SSM scan. One thread per channel d; 16 states in registers.
// B/C (shared across channels) staged in LDS in 16-timestep chunks.
// Fuses softplus(dt), y = C.h + D*x, and y*silu(z).  ygate may alias xbin
// (strict read-then-write of the same element by the same thread).
// ---------------------------------------------------------------------------
__global__ __launch_bounds__(128) void scan_kernel(const float* __restrict__ xp,
                                                   const float* __restrict__ xbin,
                                                   const float* __restrict__ xz,
                                                   const float* __restrict__ A_log,
                                                   const float* __restrict__ D_skip,
                                                   float* __restrict__ ygate) {
  const int d = blockIdx.x * 128 + threadIdx.x;   // 12 blocks * 128 = 1536
  float h[DS], A[DS];
#pragma unroll
  for (int n = 0; n < DS; ++n) {
    h[n] = 0.f;
    A[n] = -__expf(A_log[d * DS + n]);
  }
  const float Dd = D_skip[d];

  __shared__ float sB[16][DS];
  __shared__ float sC[16][DS];

  for (int tc = 0; tc < L; tc += 16) {
    __syncthreads();
    for (int l = threadIdx.x; l < 256; l += 128) {
      const int tt = l >> 4, n = l & 15;
      const float* row = xp + (size_t)(tc + tt) * NXP_PAD + DI;
      sB[tt][n] = row[n];
      sC[tt][n] = row[DS + n];
    }
    __syncthreads();
    for (int tt = 0; tt < 16; ++tt) {
      const int t = tc + tt;
      const float dtr = xp[(size_t)t * NXP_PAD + d];
      const float dt  = (dtr > 20.f) ? dtr : log1pf(__expf(dtr));
      const float xt  = xbin[(size_t)t * DI + d];
      const float zt  = xz[(size_t)t * NXZ + DI + d];
      float y = Dd * xt;
#pragma unroll
      for (int n = 0; n < DS; ++n) {
        const float ab = __expf(dt * A[n]);
        h[n] = ab * h[n] + dt * sB[tt][n] * xt;
        y += h[n] * sC[tt][n];
      }
      const float gate = zt / (1.f + __expf(-zt));
      ygate[(size_t)t * DI + d] = y * gate;
    }
  }
}

// ---------------------------------------------------------------------------
// Launch
// ---------------------------------------------------------------------------
extern "C" void kernel_launch(void* const* d_in, const int* in_sizes, int n_in,
                              void* d_out, int out_size, void* d_ws, size_t ws_size,
                              hipStream_t stream) {
  const float* x       = (const float*)d_in[0];
  const float* gamma   = (const float*)d_in[1];
  const float* beta    = (const float*)d_in[2];
  const float* W_in    = (const float*)d_in[3];
  const float* conv_w  = (const float*)d_in[4];
  const float* conv_b  = (const float*)d_in[5];
  const float* A_log   = (const float*)d_in[6];
  const float* D_skip  = (const float*)d_in[7];
  const float* W_xproj = (const float*)d_in[8];
  const float* W_out   = (const float*)d_in[9];
  float* out = (float*)d_out;

  // workspace layout (floats)
  float* ws  = (float*)d_ws;
  float* xn  = ws;                          // L*DM       = 1,572,864
  float* xz  = xn  + (size_t)L * DM;        // L*NXZ      = 6,291,456
  float* xb  = xz  + (size_t)L * NXZ;       // L*DI       = 3,145,728  (later reused as ygate)
  float* wxp = xb  + (size_t)L * DI;        // DI*NXP_PAD = 2,457,600
  float* xp  = wxp + (size_t)DI * NXP_PAD;  // L*NXP_PAD  = 3,276,800

  // 1) LayerNorm
  ln_kernel<<<L, 256, 0, stream>>>(x, gamma, beta, xn);

  // 2) xz = xn @ W_in   (2048 x 768 x 3072)
  gemm_wmma_f32<<<dim3(NXZ / 64, L / 128), 256, 0, stream>>>(
      xn, W_in, xz, L, NXZ, DM, nullptr, 0);

  // 3) depthwise conv + bias + silu -> xb
  conv_silu_kernel<<<(L * DI + 255) / 256, 256, 0, stream>>>(xz, conv_w, conv_b, xb);

  // 4) pad W_xproj to 1600 cols
  pad_wxproj<<<(DI * NXP_PAD + 255) / 256, 256, 0, stream>>>(W_xproj, wxp);

  // 5) xp = xb @ W_xproj_pad   (2048 x 1536 x 1600)
  gemm_wmma_f32<<<dim3(NXP_PAD / 64, L / 128), 256, 0, stream>>>(
      xb, wxp, xp, L, NXP_PAD, DI, nullptr, 0);

  // 6) SSM scan + gating, ygate written in-place over xb
  scan_kernel<<<DI / 128, 128, 0, stream>>>(xp, xb, xz, A_log, D_skip, xb);

  // 7) out = ygate @ W_out + x   (2048 x 1536 x 768, residual epilogue)
  gemm_wmma_f32<<<dim3(DM / 64, L / 128), 256, 0, stream>>>(
      xb, W_out, out, L, DM, DI, x, DM);
}